// ODE2_8392366096836
// MI455X (gfx1250) — compile-verified
//
#include <hip/hip_runtime.h>

#define ODE_STEP   1e-4f
#define ODE_NSUB   10
#define ODE_PQS    22.2f

// One thread integrates one trajectory (3-state generator model).
// State chain is serial: 199 intervals x 10 Euler substeps, sincos per substep.
__global__ __launch_bounds__(128) void ode_swing_gfx1250(
    const float* __restrict__ x,      // (B, T, 2) : vt, phi
    const float* __restrict__ y0_in,  // (B, 3)    : delta, omega, e1q
    const float* __restrict__ s,      // (9)
    const float* __restrict__ th,     // (9)
    float* __restrict__ out,          // (B, T, 2) : p, q
    int B, int T)
{
    const int b = blockIdx.x * blockDim.x + threadIdx.x;
    if (b >= B) return;

    // Uniform parameter products (compiler emits scalar loads / SALU for these).
    const float a   = s[0] * th[0];
    const float bb  = s[1] * th[1];
    const float k2  = s[2] * th[2];
    const float c   = s[3] * th[3];
    const float H   = s[4] * th[4];
    const float D   = s[5] * th[5];
    const float Pm  = s[6] * th[6];
    const float Ef  = s[7] * th[7];
    const float T1  = s[8] * th[8];
    const float inv_den = 1.0f / (bb * c + a * a);
    const float invH    = 1.0f / H;
    const float invT1   = 1.0f / T1;

    float delta = y0_in[b * 3 + 0];
    float omega = y0_in[b * 3 + 1];
    float e1q   = y0_in[b * 3 + 2];

    const float2* __restrict__ xrow = (const float2*)(x   + (size_t)b * T * 2);
    float2*       __restrict__ orow = (float2*)      (out + (size_t)b * T * 2);

    float2 u = xrow[0];

    for (int k = 0; k < T; ++k) {
        // gfx1250 global_prefetch_b8: pull the strided per-thread line ahead of use.
        const int kpf = (k + 16 < T) ? (k + 16) : (T - 1);
        __builtin_prefetch((const void*)&xrow[kpf], 0, 1);

        // Software-pipelined load of next interval's exogenous input.
        const int kn = (k + 1 < T) ? (k + 1) : k;
        float2 u_next = xrow[kn];

        const float vt = u.x, phi = u.y;

        // Currents at the top of interval k (state == preds[k]) — shared
        // between the p/q output and the first Euler substep.
        float sn, cs;
        __sincosf(delta - phi, &sn, &cs);
        float vd  = vt * sn;
        float vq  = vt * cs;
        float emv = e1q - vq;
        float id  = (c * emv - a * vd) * inv_den;
        float iq  = (a * emv + bb * vd) * inv_den;

        // p, q output for step k (coherent 8B store per thread).
        orow[k] = make_float2(ODE_PQS * (vd * id + vq * iq),
                              ODE_PQS * (vq * id - vd * iq));

        if (k == T - 1) break;

        // 10 Euler substeps holding u fixed (matches jax.lax.scan sub loop).
        #pragma unroll
        for (int sub = 0; sub < ODE_NSUB; ++sub) {
            const float pe = vd * id + vq * iq;
            const float nd = delta + ODE_STEP * omega;
            const float nw = omega + ODE_STEP * ((Pm - pe - D * omega) * invH);
            const float ne = e1q   + ODE_STEP * ((Ef - e1q - k2 * id) * invT1);
            delta = nd; omega = nw; e1q = ne;
            if (sub < ODE_NSUB - 1) {
                __sincosf(delta - phi, &sn, &cs);
                vd  = vt * sn;
                vq  = vt * cs;
                emv = e1q - vq;
                id  = (c * emv - a * vd) * inv_den;
                iq  = (a * emv + bb * vd) * inv_den;
            }
        }

        u = u_next;
    }
}

extern "C" void kernel_launch(void* const* d_in, const int* in_sizes, int n_in,
                              void* d_out, int out_size, void* d_ws, size_t ws_size,
                              hipStream_t stream) {
    // setup_inputs order: x (B*T*2), y0 (B*3), t (T), s (9), theta (9)
    const float* x     = (const float*)d_in[0];
    const float* y0    = (const float*)d_in[1];
    const float* s     = (const float*)d_in[3];
    const float* theta = (const float*)d_in[4];
    float* out = (float*)d_out;

    const int T = in_sizes[2];          // 200
    const int B = in_sizes[1] / 3;      // 16384

    const int block = 128;              // 4 wave32 waves per workgroup
    const int grid  = (B + block - 1) / block;
    ode_swing_gfx1250<<<grid, block, 0, stream>>>(x, y0, s, theta, out, B, T);
}